// D2T_41016937676996
// MI455X (gfx1250) — compile-verified
//
#include <hip/hip_runtime.h>

typedef __attribute__((ext_vector_type(2))) float v2f;
typedef __attribute__((ext_vector_type(4))) float v4f;
typedef __attribute__((ext_vector_type(8))) float v8f;

namespace {
constexpr int NN    = 512;   // matrix dimension
constexpr int BLK   = 512;   // threads per block = 16 waves (wave32)
constexpr int WAVES = BLK / 32;
}

__global__ __launch_bounds__(BLK)
void dcenter_kernel(const float* __restrict__ D, float* __restrict__ T) {
  __shared__ float s_row[NN];        // row sums
  __shared__ float s_col[NN];        // col sums
  __shared__ v4f   s_colp[4 * 128];  // per-row-group column partials (8 KB)
  __shared__ float s_part[WAVES];    // per-wave partials for global sum
  __shared__ float s_glob;

  const int    b    = blockIdx.x;
  const size_t base = (size_t)b * NN * NN;
  const float* Db   = D + base;
  float*       Tb   = T + base;
  const int tid  = threadIdx.x;
  const int lane = tid & 31;
  const int wave = tid >> 5;
  const int cg   = tid & 127;   // column group: columns 4*cg .. 4*cg+3
  const int rg   = tid >> 7;    // row sub-group 0..3 (uniform per wave)

  // ---------- Phase A: column sums + global sum (b128 coalesced HBM stream) ----
  {
    const float* colp = Db + (size_t)rg * NN + cg * 4;
    v4f a0 = {0.f, 0.f, 0.f, 0.f};
    v4f a1 = {0.f, 0.f, 0.f, 0.f};
    for (int i = 0; i < NN; i += 8) {
      a0 += *(const v4f*)(colp + (size_t)(i + 0) * NN);
      a1 += *(const v4f*)(colp + (size_t)(i + 4) * NN);
    }
    s_colp[rg * 128 + cg] = a0 + a1;
  }
  __syncthreads();

  // combine the 4 row-group partials for this thread's column j = tid
  const float cs = s_colp[0 * 128 + (tid >> 2)][tid & 3] +
                   s_colp[1 * 128 + (tid >> 2)][tid & 3] +
                   s_colp[2 * 128 + (tid >> 2)][tid & 3] +
                   s_colp[3 * 128 + (tid >> 2)][tid & 3];
  s_col[tid] = cs;

  // block-wide reduction for the global sum (wave32 shuffles, then LDS)
  float g = cs;
#pragma unroll
  for (int off = 16; off > 0; off >>= 1)
    g += __shfl_down(g, off, 32);
  if (lane == 0) s_part[wave] = g;
  __syncthreads();
  if (tid == 0) {
    float t = 0.f;
#pragma unroll
    for (int w = 0; w < WAVES; ++w) t += s_part[w];
    s_glob = t;
  }

  // ---------- Phase B: row sums = D x ones via V_WMMA_F32_16X16X4_F32 ----------
  // A tile layout (16x4 f32): lanes 0-15 hold (M=lane, K=0..1), lanes 16-31 hold
  // (M=lane-16, K=2..3) -> one b64 load per lane per WMMA step (L2-resident).
  v2f ones; ones[0] = 1.f; ones[1] = 1.f;      // B = all ones (layout-independent)
  const int m    = lane & 15;
  const int koff = (lane >> 4) << 1;           // 0 or 2
  for (int s = wave; s < NN / 16; s += WAVES) {  // 32 stripes, 2 per wave, uniform
    const int rb = s * 16;
    const float* rp = Db + (size_t)(rb + m) * NN + koff;
    v8f acc0 = {};
    v8f acc1 = {};
    for (int k = 0; k < NN; k += 8) {
      v2f a0 = *(const v2f*)(rp + k);
      v2f a1 = *(const v2f*)(rp + k + 4);
      acc0 = __builtin_amdgcn_wmma_f32_16x16x4_f32(false, a0, false, ones,
                                                   (short)0, acc0, false, false);
      acc1 = __builtin_amdgcn_wmma_f32_16x16x4_f32(false, a1, false, ones,
                                                   (short)0, acc1, false, false);
    }
    v8f acc = acc0 + acc1;
    // C/D layout: VGPR r -> row rb+r (lanes 0-15) / rb+8+r (lanes 16-31),
    // row sum replicated across all N columns -> lanes 0 and 16 publish.
    if (lane == 0) {
#pragma unroll
      for (int r = 0; r < 8; ++r) s_row[rb + r] = acc[r];
    } else if (lane == 16) {
#pragma unroll
      for (int r = 0; r < 8; ++r) s_row[rb + 8 + r] = acc[r];
    }
  }
  __syncthreads();

  // ---------- Phase C: apply centering (b128 coalesced, NT stores) ----------
  // T = -0.5*D + 0.5*row_mean + 0.5*col_mean - 0.5*glob_mean
  const float inv   = 1.0f / NN;
  const float gmean = s_glob * inv * inv;
  v4f aj;  // per-column constants: 0.5*col_mean - 0.5*glob_mean
#pragma unroll
  for (int c = 0; c < 4; ++c)
    aj[c] = 0.5f * (s_col[cg * 4 + c] * inv - gmean);

  const float* din  = Db + (size_t)rg * NN + cg * 4;
  float*       dout = Tb + (size_t)rg * NN + cg * 4;
  for (int i = 0; i < NN; i += 4) {
    const float ri = 0.5f * inv * s_row[i + rg];  // wave-uniform LDS broadcast
    v4f d = *(const v4f*)(din + (size_t)i * NN);  // L2 hit
    v4f o;
#pragma unroll
    for (int c = 0; c < 4; ++c)
      o[c] = fmaf(-0.5f, d[c], ri + aj[c]);
    __builtin_nontemporal_store(o, (v4f*)(dout + (size_t)i * NN));
  }
}

extern "C" void kernel_launch(void* const* d_in, const int* in_sizes, int n_in,
                              void* d_out, int out_size, void* d_ws, size_t ws_size,
                              hipStream_t stream) {
  const float* D = (const float*)d_in[0];
  float*       T = (float*)d_out;
  const int batch = in_sizes[0] / (NN * NN);  // 256
  dcenter_kernel<<<batch, BLK, 0, stream>>>(D, T);
}